// Attention_3728031613427
// MI455X (gfx1250) — compile-verified
//
#include <hip/hip_runtime.h>
#include <hip/hip_bf16.h>
#include <cstdint>

typedef __attribute__((ext_vector_type(2))) float v2f;
typedef __attribute__((ext_vector_type(8))) float v8f;

#define S_LEN 2048
#define DMODEL 4096
#define NH 32
#define NKV 8
#define HEADD 128
#define BQ 64
#define BT 32

static __device__ __forceinline__ v8f wmma_f32(v2f a, v2f b, v8f c) {
  // D = A(16x4) * B(4x16) + C(16x16), fp32, wave32
  return __builtin_amdgcn_wmma_f32_16x16x4_f32(
      /*neg_a=*/false, a, /*neg_b=*/false, b,
      /*c_mod=*/(short)0, c, /*reuse_a=*/false, /*reuse_b=*/false);
}

// Async global->LDS copy of 16 bytes per lane (CDNA5, ASYNCcnt-tracked).
static __device__ __forceinline__ void async_b128(unsigned lds_off, const void* gaddr) {
  asm volatile("global_load_async_to_lds_b128 %0, %1, off"
               :: "v"(lds_off), "v"(gaddr)
               : "memory");
}
static __device__ __forceinline__ void wait_async0() {
  asm volatile("s_wait_asynccnt 0x0" ::: "memory");
}
static __device__ __forceinline__ unsigned lds_addr(const void* p) {
  // LDS aperture: generic address low 32 bits are the LDS byte offset.
  return (unsigned)(uintptr_t)p;
}

// ---------------------------------------------------------------------------
// C[M,N] = A[M,K] @ B[N,K]^T   (row-major, M%128==0, N%64==0, K%32==0)
// 256 threads = 8 waves; block tile 128x64; each wave 32x32 (2x2 WMMA tiles).
// Double-buffered LDS tiles filled by async global->LDS DMA.
// ---------------------------------------------------------------------------
__global__ __launch_bounds__(256) void gemm_nt_kernel(
    const float* __restrict__ A, const float* __restrict__ B,
    float* __restrict__ C, int M, int N, int K) {
  __shared__ float As[2][128][36];  // Kt=32, stride 36: r*36 mod 64 all distinct
  __shared__ float Bs[2][64][36];

  const int tid   = threadIdx.x;
  const int wid   = tid >> 5;
  const int lane  = tid & 31;
  const int lhalf = lane & 15;
  const int hi    = lane >> 4;
  const int koff  = hi * 2;            // K offset of this lane's v2f fragment
  const int bm    = blockIdx.y * 128;
  const int bn    = blockIdx.x * 64;
  const int wr    = (wid >> 1) * 32;   // wave row offset in block tile (0..96)
  const int wc    = (wid & 1) * 32;    // wave col offset in block tile (0,32)

  // Issue async DMA for one K-tile into buffer `buf`.
  auto issue_tile = [&](int buf, int kb) {
    // A: 128x32 floats = 1024 float4 -> 4 per thread
#pragma unroll
    for (int j = 0; j < 4; ++j) {
      int idx = tid + j * 256;
      int r   = idx >> 3;
      int c4  = idx & 7;
      async_b128(lds_addr(&As[buf][r][c4 * 4]),
                 A + (size_t)(bm + r) * K + kb + c4 * 4);
    }
    // B: 64x32 floats = 512 float4 -> 2 per thread
#pragma unroll
    for (int j = 0; j < 2; ++j) {
      int idx = tid + j * 256;
      int r   = idx >> 3;
      int c4  = idx & 7;
      async_b128(lds_addr(&Bs[buf][r][c4 * 4]),
                 B + (size_t)(bn + r) * K + kb + c4 * 4);
    }
  };

  v8f zero8 = {};
  v8f acc[2][2];
  acc[0][0] = zero8; acc[0][1] = zero8; acc[1][0] = zero8; acc[1][1] = zero8;

  const int nt = K / 32;
  issue_tile(0, 0);
  for (int i = 0; i < nt; ++i) {
    wait_async0();        // this wave's tile-i DMA complete
    __syncthreads();      // all waves' tile-i DMA complete; prev reads retired
    if (i + 1 < nt) issue_tile((i + 1) & 1, (i + 1) * 32);  // overlap with WMMA
    const int b = i & 1;
#pragma unroll
    for (int ks = 0; ks < 8; ++ks) {
      const int k0 = ks * 4 + koff;
      v2f a0 = *(const v2f*)&As[b][wr + lhalf][k0];
      v2f a1 = *(const v2f*)&As[b][wr + 16 + lhalf][k0];
      v2f b0 = *(const v2f*)&Bs[b][wc + lhalf][k0];
      v2f b1 = *(const v2f*)&Bs[b][wc + 16 + lhalf][k0];
      acc[0][0] = wmma_f32(a0, b0, acc[0][0]);
      acc[0][1] = wmma_f32(a0, b1, acc[0][1]);
      acc[1][0] = wmma_f32(a1, b0, acc[1][0]);
      acc[1][1] = wmma_f32(a1, b1, acc[1][1]);
    }
  }

  // C/D layout: VGPR v -> row v (lanes 0-15) / v+8 (lanes 16-31), col = lane%16
#pragma unroll
  for (int i = 0; i < 2; ++i)
#pragma unroll
    for (int jj = 0; jj < 2; ++jj)
#pragma unroll
      for (int v = 0; v < 8; ++v) {
        int row = bm + wr + i * 16 + v + hi * 8;
        int col = bn + wc + jj * 16 + lhalf;
        C[(size_t)row * N + col] = acc[i][jj][v];
      }
}

// ---------------------------------------------------------------------------
// RoPE (interleaved pairs), in place.  x: [S, nheads, 128]
// ---------------------------------------------------------------------------
__global__ void rope_kernel(float* __restrict__ x,
                            const float* __restrict__ cosT,
                            const float* __restrict__ sinT, int nheads) {
  int idx = blockIdx.x * blockDim.x + threadIdx.x;
  int total = S_LEN * nheads * (HEADD / 2);
  if (idx >= total) return;
  int i = idx & 63;
  int h = (idx >> 6) % nheads;
  int s = idx / (64 * nheads);
  float c  = cosT[s * 64 + i];
  float sn = sinT[s * 64 + i];
  float* p = x + ((size_t)s * nheads + h) * HEADD + 2 * i;
  float xr = p[0], xi = p[1];
  p[0] = xr * c - xi * sn;
  p[1] = xr * sn + xi * c;
}

// ---------------------------------------------------------------------------
// Flash attention, causal, GQA (4 q-heads per kv-head).
// grid = (S/64, H); 128 threads = 4 waves; wave w owns q rows [qb*64+w*16, +16)
// K/V tiles staged with async global->LDS DMA.
// ---------------------------------------------------------------------------
__global__ __launch_bounds__(128) void flash_attn_kernel(
    const float* __restrict__ xq, const float* __restrict__ xk,
    const float* __restrict__ xv, float* __restrict__ attn) {
  __shared__ float Ks[BT][HEADD + 4];   // 32x132: r*132 mod 64 spreads banks
  __shared__ float Vs[BT][HEADD + 4];
  __shared__ float Ps[4][16][BT + 4];   // per-wave P staging (C->A relayout)

  const int qb    = blockIdx.x;
  const int h     = blockIdx.y;
  const int kv    = h >> 2;             // REP = 4
  const int tid   = threadIdx.x;
  const int wid   = tid >> 5;
  const int lane  = tid & 31;
  const int lhalf = lane & 15;
  const int hi    = lane >> 4;
  const int koff  = hi * 2;

  const int q0   = qb * BQ + wid * 16;  // wave's first q row
  const int qrow = q0 + lhalf;          // this lane's A-operand row

  // Preload Q fragments (A-operand) into registers: 32 v2f covering d=0..127
  v2f qreg[32];
  const float* qptr = xq + ((size_t)qrow * NH + h) * HEADD + koff;
#pragma unroll
  for (int ks = 0; ks < 32; ++ks) qreg[ks] = *(const v2f*)(qptr + ks * 4);

  v8f zero8 = {};
  v8f o[8];
#pragma unroll
  for (int dt = 0; dt < 8; ++dt) o[dt] = zero8;
  float rowm[8], rowl[8];
#pragma unroll
  for (int v = 0; v < 8; ++v) { rowm[v] = -1e30f; rowl[v] = 0.0f; }

  const float scale = 0.08838834764831845f;  // 1/sqrt(128)
  const int ntb = 2 * qb + 2;                // t-blocks of 32 in causal span

  for (int tb = 0; tb < ntb; ++tb) {
    const int t0 = tb * BT;
    // Async K/V tile DMA: 32x128 each = 1024 float4; 8 per thread per matrix
#pragma unroll
    for (int j = 0; j < 8; ++j) {
      int idx = tid + j * 128;
      int r   = idx >> 5;
      int c4  = idx & 31;
      size_t g = ((size_t)(t0 + r) * NKV + kv) * HEADD + c4 * 4;
      async_b128(lds_addr(&Ks[r][c4 * 4]), xk + g);
      async_b128(lds_addr(&Vs[r][c4 * 4]), xv + g);
    }
    wait_async0();
    __syncthreads();

    const bool skip = (t0 > q0 + 15);   // wave-uniform: whole t-block masked
    if (!skip) {
      // S = Q K^T  -> two 16x16 tiles over t
      v8f st0 = zero8, st1 = zero8;
#pragma unroll
      for (int ks = 0; ks < 32; ++ks) {
        const int k0 = ks * 4 + koff;
        v2f b0 = *(const v2f*)&Ks[lhalf][k0];
        v2f b1 = *(const v2f*)&Ks[lhalf + 16][k0];
        st0 = wmma_f32(qreg[ks], b0, st0);
        st1 = wmma_f32(qreg[ks], b1, st1);
      }
      const bool needMask = (t0 + BT - 1 > q0);
      // Online softmax update (row stats redundant across each 16-lane half)
#pragma unroll
      for (int v = 0; v < 8; ++v) {
        const int qg = q0 + v + hi * 8;
        float s0 = st0[v] * scale;
        float s1 = st1[v] * scale;
        if (needMask) {
          if (t0 + lhalf > qg)      s0 = -1e30f;
          if (t0 + 16 + lhalf > qg) s1 = -1e30f;
        }
        float mx = fmaxf(s0, s1);
#pragma unroll
        for (int d = 1; d < 16; d <<= 1) mx = fmaxf(mx, __shfl_xor(mx, d, 32));
        float mnew = fmaxf(rowm[v], mx);
        float p0 = __expf(s0 - mnew);
        float p1 = __expf(s1 - mnew);
        float rs = p0 + p1;
#pragma unroll
        for (int d = 1; d < 16; d <<= 1) rs += __shfl_xor(rs, d, 32);
        float alpha = __expf(rowm[v] - mnew);
        rowm[v] = mnew;
        rowl[v] = rowl[v] * alpha + rs;
#pragma unroll
        for (int dt = 0; dt < 8; ++dt) o[dt][v] = o[dt][v] * alpha;
        // C-layout -> LDS (A-layout readback after barrier)
        Ps[wid][v + hi * 8][lhalf]      = p0;
        Ps[wid][v + hi * 8][16 + lhalf] = p1;
      }
    }
    __syncthreads();   // P visible; Ks reads done

    if (!skip) {
      // O += P V : P[16,32] (A-operand from LDS) x V[32,128]
#pragma unroll
      for (int ks = 0; ks < 8; ++ks) {
        const int k0 = ks * 4 + koff;
        v2f a = *(const v2f*)&Ps[wid][lhalf][k0];
#pragma unroll
        for (int dt = 0; dt < 8; ++dt) {
          v2f b;
          b.x = Vs[k0][dt * 16 + lhalf];
          b.y = Vs[k0 + 1][dt * 16 + lhalf];
          o[dt] = wmma_f32(a, b, o[dt]);
        }
      }
    }
    __syncthreads();   // Vs reads done before next tile's DMA overwrites
  }

  // Normalize and store: attn[s, h, d]
#pragma unroll
  for (int v = 0; v < 8; ++v) {
    const int qg = q0 + v + hi * 8;
    float inv = 1.0f / rowl[v];
#pragma unroll
    for (int dt = 0; dt < 8; ++dt) {
      attn[((size_t)qg * NH + h) * HEADD + dt * 16 + lhalf] = o[dt][v] * inv;
    }
  }
}

// ---------------------------------------------------------------------------
// Host orchestration
// ---------------------------------------------------------------------------
extern "C" void kernel_launch(void* const* d_in, const int* in_sizes, int n_in,
                              void* d_out, int out_size, void* d_ws, size_t ws_size,
                              hipStream_t stream) {
  const float* x    = (const float*)d_in[0];
  const float* wq   = (const float*)d_in[1];
  const float* wk   = (const float*)d_in[2];
  const float* wv   = (const float*)d_in[3];
  const float* wo   = (const float*)d_in[4];
  const float* cosT = (const float*)d_in[5];
  const float* sinT = (const float*)d_in[6];
  // d_in[7]=cache, d_in[8]=mask, d_in[9]=start_pos: not needed (start_pos==0,
  // causal mask applied analytically, cache is not part of the output).
  (void)in_sizes; (void)n_in; (void)out_size; (void)ws_size;

  float* out  = (float*)d_out;
  float* xqb  = (float*)d_ws;                          // [2048, 32,128] 32 MB
  float* xkb  = xqb + (size_t)S_LEN * NH * HEADD;      // [2048,  8,128]  8 MB
  float* xvb  = xkb + (size_t)S_LEN * NKV * HEADD;     // [2048,  8,128]  8 MB
  float* attn = xvb + (size_t)S_LEN * NKV * HEADD;     // [2048, 32,128] 32 MB

  dim3 gblk(256);
  // QKV projections: X[2048,4096] @ W^T
  gemm_nt_kernel<<<dim3(DMODEL / 64, S_LEN / 128), gblk, 0, stream>>>(
      x, wq, xqb, S_LEN, NH * HEADD, DMODEL);
  gemm_nt_kernel<<<dim3((NKV * HEADD) / 64, S_LEN / 128), gblk, 0, stream>>>(
      x, wk, xkb, S_LEN, NKV * HEADD, DMODEL);
  gemm_nt_kernel<<<dim3((NKV * HEADD) / 64, S_LEN / 128), gblk, 0, stream>>>(
      x, wv, xvb, S_LEN, NKV * HEADD, DMODEL);

  // RoPE on Q and K
  rope_kernel<<<(S_LEN * NH * (HEADD / 2)) / 256, 256, 0, stream>>>(xqb, cosT, sinT, NH);
  rope_kernel<<<(S_LEN * NKV * (HEADD / 2)) / 256, 256, 0, stream>>>(xkb, cosT, sinT, NKV);

  // Causal GQA flash attention
  flash_attn_kernel<<<dim3(S_LEN / 64, NH), dim3(128), 0, stream>>>(xqb, xkb, xvb, attn);

  // Output projection
  gemm_nt_kernel<<<dim3(DMODEL / 64, S_LEN / 128), gblk, 0, stream>>>(
      attn, wo, out, S_LEN, DMODEL, DMODEL);
}